// ThreatIntelDB_80118319939817
// MI455X (gfx1250) — compile-verified
//
#include <hip/hip_runtime.h>
#include <math.h>

// ---------------- problem constants (from reference) ----------------
#define DIMS     1024        // D
#define NEMB     1024        // N embeddings (sequential steps)
#define BROWS    8192        // bank rows
#define CHUNK    16          // embeddings per GEMM chunk
#define NCHUNK   (NEMB / CHUNK)

// ---------------- machine mapping ----------------
#define NUM_WG       128     // one WGP-resident persistent workgroup each
#define ROWS_PER_WG  (BROWS / NUM_WG)   // 64 rows owned per WG
#define LDSTRIDE     1028    // 1024 + 4 pad -> bank = (4*row + k) % 64, conflict-free
#define BLOCK        256     // 8 wave32: 4 tiles x 2 K-halves

typedef __attribute__((ext_vector_type(2))) float v2f;
typedef __attribute__((ext_vector_type(4))) float v4f;
typedef __attribute__((ext_vector_type(8))) float v8f;

// -------- order-preserving float <-> uint key (for packed argmax) --------
__device__ __forceinline__ unsigned f2key(float f) {
    unsigned u = __float_as_uint(f);
    return (u & 0x80000000u) ? ~u : (u | 0x80000000u);
}
__device__ __forceinline__ float key2f(unsigned k) {
    unsigned u = (k & 0x80000000u) ? (k & 0x7FFFFFFFu) : ~k;
    return __uint_as_float(u);
}

// -------- phase-indexed grid barrier (counters pre-zeroed by init kernel) --------
__device__ __forceinline__ void gridBarrier(unsigned* bars, int phase) {
    __syncthreads();
    if (threadIdx.x == 0) {
        __threadfence();
        atomicAdd(&bars[phase], 1u);
        while (__hip_atomic_load(&bars[phase], __ATOMIC_ACQUIRE,
                                 __HIP_MEMORY_SCOPE_AGENT) < NUM_WG) {
            __builtin_amdgcn_s_sleep(2);
        }
    }
    __syncthreads();
}

// ================= init kernels (run before the persistent kernel) =================
__global__ void k_zero_ctrl(unsigned long long* slots, unsigned* bars) {
    int t = blockIdx.x * blockDim.x + threadIdx.x;
    if (t < NEMB) { slots[t] = 0ull; bars[t] = 0u; }
}

__global__ void k_copy_state(const float* __restrict__ bankIn,
                             const float* __restrict__ cntIn,
                             float* __restrict__ bankOut,
                             float* __restrict__ cntOut) {
    size_t n = (size_t)BROWS * DIMS;
    size_t stride = (size_t)gridDim.x * blockDim.x;
    for (size_t t = (size_t)blockIdx.x * blockDim.x + threadIdx.x; t < n; t += stride)
        bankOut[t] = bankIn[t];
    for (size_t t = (size_t)blockIdx.x * blockDim.x + threadIdx.x; t < BROWS; t += stride)
        cntOut[t] = cntIn[t];
}

// one block per embedding row: emb_n = emb / max(||emb||, eps)
__global__ void k_norm_emb(const float* __restrict__ emb, float* __restrict__ embN) {
    __shared__ float red[256];
    const int row = blockIdx.x;
    const float* src = emb + (size_t)row * DIMS;
    float p = 0.f;
    for (int d = threadIdx.x; d < DIMS; d += 256) { float v = src[d]; p += v * v; }
    red[threadIdx.x] = p; __syncthreads();
    for (int s = 128; s > 0; s >>= 1) {
        if (threadIdx.x < s) red[threadIdx.x] += red[threadIdx.x + s];
        __syncthreads();
    }
    float inv = 1.0f / fmaxf(sqrtf(red[0]), 1e-12f);
    for (int d = threadIdx.x; d < DIMS; d += 256)
        embN[(size_t)row * DIMS + d] = src[d] * inv;
}

// ================= persistent sequential-ingest kernel =================
__global__ __launch_bounds__(BLOCK)
void k_ingest(const float* __restrict__ embIn,   // raw embeddings (EMA uses raw)
              const float* __restrict__ embN,    // normalized embeddings (d_ws)
              float* bankOut, float* cntOut,     // in-place state (d_out)
              unsigned long long* slots,         // NEMB packed argmax slots
              unsigned* bars,                    // NEMB barrier counters
              const int* ptrIn, const int* genIn,
              float* ptrOut) {
    extern __shared__ float smem[];
    float* sBank = smem;                                  // 64 x 1028 normalized rows
    float* sS    = sBank + ROWS_PER_WG * LDSTRIDE;        // 64 x 16 sims (waves 0-3)
    float* sS2   = sS + ROWS_PER_WG * CHUNK;              // 64 x 16 partial (waves 4-7)
    float* sRed  = sS2 + ROWS_PER_WG * CHUNK;             // BLOCK reduction scratch
    __shared__ unsigned long long sKeys[ROWS_PER_WG];
    __shared__ unsigned long long sBcast;

    const int tid = threadIdx.x;
    const int wg = blockIdx.x;
    const int rowBase = wg * ROWS_PER_WG;

    // effective novelty threshold (generation is a runtime scalar input)
    const float gen = (float)genIn[0];
    const float alpha = fminf(gen / 50.0f, 1.0f);
    const float simCut = 1.0f - ((1.0f - alpha) + 0.1f * alpha);
    int ptrLocal = ptrIn[0];

    // ---- Phase A: async DMA this WG's bank slice global->LDS, then normalize ----
    // GLOBAL_LOAD_ASYNC_TO_LDS_B128: per lane 16B, 256 threads = one 4KB row per
    // issue; ASYNCcnt-tracked (issue stalls safely if >63 outstanding). LDS dest
    // address = low 32 bits of the flat shared pointer (ISA aperture rule).
    {
        unsigned ldsOff = (unsigned)(size_t)sBank + (unsigned)(tid * 16);
        const float* g = bankOut + (size_t)rowBase * DIMS + tid * 4;
        for (int lr = 0; lr < ROWS_PER_WG; ++lr) {
            asm volatile("global_load_async_to_lds_b128 %0, %1, off"
                         :: "v"(ldsOff), "v"(g) : "memory");
            ldsOff += LDSTRIDE * 4;   // padded LDS row stride (4112 B)
            g += DIMS;                // dense global row stride (4096 B)
        }
        asm volatile("s_wait_asynccnt 0x0" ::: "memory");
        __syncthreads();
        for (int lr = 0; lr < ROWS_PER_WG; ++lr) {
            float p = 0.f;
            for (int d = tid; d < DIMS; d += BLOCK) {
                float v = sBank[lr * LDSTRIDE + d];
                p += v * v;
            }
            sRed[tid] = p; __syncthreads();
            for (int s = BLOCK / 2; s > 0; s >>= 1) {
                if (tid < s) sRed[tid] += sRed[tid + s];
                __syncthreads();
            }
            float inv = 1.0f / fmaxf(sqrtf(sRed[0]), 1e-12f);
            __syncthreads();
            for (int d = tid; d < DIMS; d += BLOCK) sBank[lr * LDSTRIDE + d] *= inv;
            __syncthreads();
        }
    }

    int step = 0;
    for (int c = 0; c < NCHUNK; ++c) {
        // warm L2 lines for this chunk's B-matrix (emb_n rows)
        {
            const float* chunkB = embN + (size_t)c * CHUNK * DIMS;
            for (int off = tid * 16; off < CHUNK * DIMS; off += BLOCK * 16)
                __builtin_prefetch(chunkB + off, 0, 1);
        }

        // ---- Phase B: S(64x16) = sBank(64x1024) @ embN_chunk^T via f32 WMMA ----
        // 8 waves: wave w computes tile (w&3) over K slab (w>>2)*512.
        // Contiguous-K remap: within a wave's 512-wide slab, lane-half kh owns the
        // contiguous 256-float range kh*256.. -> b128 loads for both A and B.
        {
            const int lane = tid & 31;
            const int wv   = tid >> 5;
            const int tile = wv & 3;          // which 16-row tile
            const int ks   = wv >> 2;         // which 512-wide K slab
            const int m    = lane & 15;       // A row-in-tile; B column (embedding)
            const int kh   = lane >> 4;       // 256-float sub-range within slab
            const int tr   = tile * 16;
            const float* aP = sBank + (size_t)(tr + m) * LDSTRIDE + ks * 512 + kh * 256;
            const float* bP = embN + (size_t)(c * CHUNK + m) * DIMS + ks * 512 + kh * 256;
            v8f acc0 = {}, acc1 = {}, acc2 = {}, acc3 = {};
#pragma unroll 4
            for (int k = 0; k < 256; k += 8) {
                v4f a01 = *(const v4f*)(aP + k);
                v4f a23 = *(const v4f*)(aP + k + 4);
                v4f b01 = *(const v4f*)(bP + k);
                v4f b23 = *(const v4f*)(bP + k + 4);
                v2f aa, bb;
                aa[0] = a01.x; aa[1] = a01.y; bb[0] = b01.x; bb[1] = b01.y;
                acc0 = __builtin_amdgcn_wmma_f32_16x16x4_f32(false, aa, false, bb,
                                                             (short)0, acc0, false, false);
                aa[0] = a01.z; aa[1] = a01.w; bb[0] = b01.z; bb[1] = b01.w;
                acc1 = __builtin_amdgcn_wmma_f32_16x16x4_f32(false, aa, false, bb,
                                                             (short)0, acc1, false, false);
                aa[0] = a23.x; aa[1] = a23.y; bb[0] = b23.x; bb[1] = b23.y;
                acc2 = __builtin_amdgcn_wmma_f32_16x16x4_f32(false, aa, false, bb,
                                                             (short)0, acc2, false, false);
                aa[0] = a23.z; aa[1] = a23.w; bb[0] = b23.z; bb[1] = b23.w;
                acc3 = __builtin_amdgcn_wmma_f32_16x16x4_f32(false, aa, false, bb,
                                                             (short)0, acc3, false, false);
            }
            v8f acc = (acc0 + acc1) + (acc2 + acc3);
            __syncthreads();                  // sS/sS2 reuse from previous chunk
            // C layout: vgpr v -> M = v + 8*kh ; N = lane&15
            float* dst = (ks == 0) ? sS : sS2;
            for (int v = 0; v < 8; ++v)
                dst[(tr + v + 8 * kh) * CHUNK + m] = acc[v];
            __syncthreads();
            // combine the two K-slab partials
            for (int t = tid; t < ROWS_PER_WG * CHUNK; t += BLOCK)
                sS[t] += sS2[t];
            __syncthreads();
        }

        // ---- Phase C: 16 sequential steps, exact via one-row lazy fixups ----
        for (int i = 0; i < CHUNK; ++i, ++step) {
            // per-WG argmax over our 64 rows of column i
            if (tid < ROWS_PER_WG) {
                unsigned k = f2key(sS[tid * CHUNK + i]);
                sKeys[tid] = ((unsigned long long)k << 32) |
                             (unsigned long long)(0xFFFFFFFFu - (unsigned)(rowBase + tid));
            }
            __syncthreads();
            for (int s = ROWS_PER_WG / 2; s > 0; s >>= 1) {
                if (tid < s && sKeys[tid + s] > sKeys[tid]) sKeys[tid] = sKeys[tid + s];
                __syncthreads();
            }
            if (tid == 0) atomicMax(&slots[step], sKeys[0]);

            gridBarrier(bars, step);

            if (tid == 0)
                sBcast = __hip_atomic_load(&slots[step], __ATOMIC_RELAXED,
                                           __HIP_MEMORY_SCOPE_AGENT);
            __syncthreads();
            const unsigned long long kk = sBcast;
            const float maxsim = key2f((unsigned)(kk >> 32));
            const int maxrow = (int)(0xFFFFFFFFu - (unsigned)(kk & 0xFFFFFFFFu));
            const bool dup = maxsim > simCut;
            const int idx = dup ? maxrow : (ptrLocal % BROWS);
            if (!dup) ptrLocal++;              // every WG tracks ptr identically

            if ((idx >> 6) == wg) {            // block-uniform branch: we own this row
                const int lr = idx & (ROWS_PER_WG - 1);
                const int ge = c * CHUNK + i;
                float p = 0.f;
                for (int d = tid; d < DIMS; d += BLOCK) {
                    float bv = bankOut[(size_t)idx * DIMS + d];
                    float ev = embIn[(size_t)ge * DIMS + d];
                    float nv = dup ? (0.995f * bv + 0.005f * ev) : ev;
                    bankOut[(size_t)idx * DIMS + d] = nv;
                    sBank[lr * LDSTRIDE + d] = nv;     // unnormalized for now
                    p += nv * nv;
                }
                sRed[tid] = p; __syncthreads();
                for (int s = BLOCK / 2; s > 0; s >>= 1) {
                    if (tid < s) sRed[tid] += sRed[tid + s];
                    __syncthreads();
                }
                float inv = 1.0f / fmaxf(sqrtf(sRed[0]), 1e-12f);
                __syncthreads();
                for (int d = tid; d < DIMS; d += BLOCK)
                    sBank[lr * LDSTRIDE + d] *= inv;   // now normalized
                __syncthreads();
                // fix up this row's sims vs remaining chunk embeddings
                for (int j = i + 1; j < CHUNK; ++j) {
                    const float* en = embN + (size_t)(c * CHUNK + j) * DIMS;
                    float q = 0.f;
                    for (int d = tid; d < DIMS; d += BLOCK)
                        q += sBank[lr * LDSTRIDE + d] * en[d];
                    sRed[tid] = q; __syncthreads();
                    for (int s = BLOCK / 2; s > 0; s >>= 1) {
                        if (tid < s) sRed[tid] += sRed[tid + s];
                        __syncthreads();
                    }
                    if (tid == 0) sS[lr * CHUNK + j] = sRed[0];
                    __syncthreads();
                }
                if (tid == 0) cntOut[idx] = dup ? cntOut[idx] + 1.0f : 1.0f;
            }
            __syncthreads();
        }
    }

    if (wg == 0 && tid == 0) *ptrOut = (float)ptrLocal;
}

// ================= host entry =================
extern "C" void kernel_launch(void* const* d_in, const int* in_sizes, int n_in,
                              void* d_out, int out_size, void* d_ws, size_t ws_size,
                              hipStream_t stream) {
    const float* embIn  = (const float*)d_in[0];   // [1024,1024] f32
    const float* bankIn = (const float*)d_in[1];   // [8192,1024] f32
    const float* cntIn  = (const float*)d_in[2];   // [8192] f32
    const int*   ptrIn  = (const int*)d_in[3];     // scalar
    const int*   genIn  = (const int*)d_in[4];     // scalar

    float* bankOut = (float*)d_out;                       // [8192,1024]
    float* cntOut  = bankOut + (size_t)BROWS * DIMS;      // [8192]
    float* ptrOut  = cntOut + BROWS;                      // scalar

    // workspace: emb_n (4MB) | argmax slots (8KB) | barrier counters (4KB)
    float* embN = (float*)d_ws;
    unsigned long long* slots =
        (unsigned long long*)((char*)d_ws + (size_t)NEMB * DIMS * sizeof(float));
    unsigned* bars = (unsigned*)(slots + NEMB);

    k_zero_ctrl<<<(NEMB + 255) / 256, 256, 0, stream>>>(slots, bars);
    k_copy_state<<<1024, 256, 0, stream>>>(bankIn, cntIn, bankOut, cntOut);
    k_norm_emb<<<NEMB, 256, 0, stream>>>(embIn, embN);

    const size_t ldsBytes =
        (ROWS_PER_WG * LDSTRIDE + 2 * ROWS_PER_WG * CHUNK + BLOCK) * sizeof(float);
    k_ingest<<<NUM_WG, BLOCK, ldsBytes, stream>>>(
        embIn, embN, bankOut, cntOut, slots, bars, ptrIn, genIn, ptrOut);
}